// AttentionLayer_10995116278518
// MI455X (gfx1250) — compile-verified
//
#include <hip/hip_runtime.h>

// Problem constants (from reference)
#define SEQ   512
#define BATCH 64
#define EMB   300
#define FEAT  4096
#define NROW  (SEQ * BATCH)   // 32768 rows, row r = s*BATCH + b
#define NCHUNK 5              // 1 emb chunk + 4 lstm quarters

typedef float v2f __attribute__((ext_vector_type(2)));
typedef float v4f __attribute__((ext_vector_type(4)));
typedef float v8f __attribute__((ext_vector_type(8)));

// D = A(16x4) * B(4x16) + C, fp32. A = attn broadcast (all 16 rows equal),
// B columns = 16 data rows => C[0][n] (lanes 0..15, VGPR0) = dot for row n.
__device__ __forceinline__ v8f wmma_acc(v2f a, v2f b, v8f c) {
  return __builtin_amdgcn_wmma_f32_16x16x4_f32(
      /*neg_a=*/false, a, /*neg_b=*/false, b,
      /*c_mod=*/(short)0, c, /*reuse_a=*/false, /*reuse_b=*/false);
}

// Grid: 2048 blocks (one 16-row group each) x 160 threads (5 waves).
// Wave 0: embedded[0:300); waves 1..4: lstm quarter (w-1)*1024 .. +1024.
// Lane mapping (ISA 16x16x4 layout): lanes 0-15 hold K=0,1; lanes 16-31 K=2,3.
// Lane t handles row n = t&15; hi = t>>4 selects the K=hi*4..+3 half, so a
// per-lane float4 load at feature f + hi*4 feeds two WMMA fragments (K=8/iter).
__global__ __launch_bounds__(160) void attn_partial_dot(
    const float* __restrict__ emb, const float* __restrict__ lstm,
    const float* __restrict__ attn, float* __restrict__ ws) {
  const int t    = threadIdx.x;
  const int wave = t >> 5;        // chunk id 0..4 (wave-uniform: EXEC stays full)
  const int lane = t & 31;
  const int n    = lane & 15;
  const int hi   = lane >> 4;     // 0: K=0..1 slots, 1: K=2..3 slots
  const int r    = blockIdx.x * 16 + n;  // data row for this lane (both halves)

  v8f c0 = {};
  v8f c1 = {};

  if (wave == 0) {
    // ---- embedded chunk: 300 features = 37 * K8 + 1 * K4 tail ----
    const float* rowp = emb  + (size_t)r * EMB + hi * 4;
    const float* ap   = attn + hi * 4;                 // attn[0:300)
#pragma unroll 4
    for (int it = 0; it < 37; ++it) {
      v4f d = *reinterpret_cast<const v4f*>(rowp);     // global_load_b128
      v4f a = *reinterpret_cast<const v4f*>(ap);       // hot in cache (17.6 KB)
      c0 = wmma_acc(a.lo, d.lo, c0);
      c1 = wmma_acc(a.hi, d.hi, c1);
      rowp += 8; ap += 8;
    }
    {
      // tail: features 296..299 (K=4), float2 per lane
      v2f d = *reinterpret_cast<const v2f*>(emb  + (size_t)r * EMB + 296 + hi * 2);
      v2f a = *reinterpret_cast<const v2f*>(attn + 296 + hi * 2);
      c0 = wmma_acc(a, d, c0);
    }
  } else {
    // ---- lstm quarter: 1024 features = 128 * K8 ----
    const int base = (wave - 1) * 1024;
    const float* rowp = lstm + (size_t)r * FEAT + base + hi * 4;
    const float* ap   = attn + EMB + base + hi * 4;    // attn[300:4396)
#pragma unroll 4
    for (int it = 0; it < 128; ++it) {
      v4f d = *reinterpret_cast<const v4f*>(rowp);
      v4f a = *reinterpret_cast<const v4f*>(ap);
      c0 = wmma_acc(a.lo, d.lo, c0);
      c1 = wmma_acc(a.hi, d.hi, c1);
      rowp += 8; ap += 8;
    }
  }

  // C[0][n] lives in accumulator VGPR0, lanes 0..15 (lanes 16..31 redundant).
  if (hi == 0) {
    const float partial = c0[0] + c1[0];
    const int b = r & (BATCH - 1);
    const int s = r >> 6;  // r / BATCH
    // ws layout: [chunk][b][s] -> contiguous-s reads in the softmax pass.
    ws[(size_t)wave * NROW + (size_t)b * SEQ + s] = partial;
  }
}

// 64 blocks (one per batch row) x 512 threads (one per seq pos).
// Sums the 5 partial slices, then stable softmax over s.
__global__ __launch_bounds__(512) void attn_softmax(
    const float* __restrict__ ws, float* __restrict__ out) {
  const int b = blockIdx.x;
  const int s = threadIdx.x;

  float v = 0.f;
#pragma unroll
  for (int c = 0; c < NCHUNK; ++c)
    v += ws[(size_t)c * NROW + (size_t)b * SEQ + s];

  __shared__ float sm[SEQ];
  sm[s] = v;
  __syncthreads();
  for (int off = SEQ / 2; off > 0; off >>= 1) {
    if (s < off) sm[s] = fmaxf(sm[s], sm[s + off]);
    __syncthreads();
  }
  const float m = sm[0];
  __syncthreads();

  const float e = expf(v - m);
  sm[s] = e;
  __syncthreads();
  for (int off = SEQ / 2; off > 0; off >>= 1) {
    if (s < off) sm[s] += sm[s + off];
    __syncthreads();
  }
  out[(size_t)b * SEQ + s] = e / sm[0];
}

extern "C" void kernel_launch(void* const* d_in, const int* in_sizes, int n_in,
                              void* d_out, int out_size, void* d_ws, size_t ws_size,
                              hipStream_t stream) {
  const float* embedded = (const float*)d_in[0];  // [512,64,300] f32
  const float* lstm     = (const float*)d_in[1];  // [512,64,4096] f32
  const float* attn     = (const float*)d_in[2];  // [4396] f32
  // d_in[3] = mask: dead code in the reference (masked_fill result discarded).
  float* out = (float*)d_out;                     // [64,512] f32
  float* ws  = (float*)d_ws;                      // needs 5*32768*4 = 640 KB

  attn_partial_dot<<<dim3(NROW / 16), dim3(160), 0, stream>>>(embedded, lstm, attn, ws);
  attn_softmax<<<dim3(BATCH), dim3(SEQ), 0, stream>>>(ws, out);
}